// GAT_53386443489832
// MI455X (gfx1250) — compile-verified
//
#include <hip/hip_runtime.h>
#include <hip/hip_bf16.h>

typedef __attribute__((ext_vector_type(2))) float v2f;
typedef __attribute__((ext_vector_type(8))) float v8f;

#define NNODES 50000
#define NEDGES 800000
#define E2     850000      // edges + self loops
#define HC     128         // H*C
#define NHEAD  2
#define NCLS   10
#define NGRAPH 64
#define NEG_SLOPE 0.2f

// ---------------------------------------------------------------------------
// GEMM: Hout[N,128] = X[N,128] @ Wm[128,128] via V_WMMA_F32_16X16X4_F32.
// Block = 256 threads = 8 waves; wave w owns column tile w, block owns one
// 16-row tile. 32 k-steps of exact-fp32 WMMA. (verified: emits v_wmma)
// ---------------------------------------------------------------------------
__global__ __launch_bounds__(256) void gat_gemm_wmma(
    const float* __restrict__ X, const float* __restrict__ Wm,
    float* __restrict__ Hout, int nrows)
{
    const int wave  = threadIdx.x >> 5;
    const int lane  = threadIdx.x & 31;
    const int lhalf = lane >> 4;
    const int l15   = lane & 15;
    const int row0  = blockIdx.x * 16;
    const int col0  = wave * 16;
    if (row0 >= nrows) return;

    v8f acc = {};
    const float* arow = X + (size_t)(row0 + l15) * HC;
    #pragma unroll 8
    for (int k0 = 0; k0 < HC; k0 += 4) {
        const int ka = k0 + 2 * lhalf;
        v2f a; a.x = arow[ka];               a.y = arow[ka + 1];
        v2f b; b.x = Wm[(size_t)ka * HC + col0 + l15];
               b.y = Wm[(size_t)(ka + 1) * HC + col0 + l15];
        acc = __builtin_amdgcn_wmma_f32_16x16x4_f32(
                  false, a, false, b, (short)0, acc, false, false);
    }
    float* orow = Hout + (size_t)(row0 + 8 * lhalf) * HC + col0 + l15;
    #pragma unroll
    for (int r = 0; r < 8; ++r)
        orow[(size_t)r * HC] = acc[r];
}

// ---------------------------------------------------------------------------
// Per-node attention logits: one wave32 per node, shuffle reduction.
// ---------------------------------------------------------------------------
__global__ void gat_alpha(const float* __restrict__ Hb,
                          const float* __restrict__ as_, const float* __restrict__ ad_,
                          float* __restrict__ oas, float* __restrict__ oad, int n)
{
    const int node = blockIdx.x * (blockDim.x >> 5) + (threadIdx.x >> 5);
    const int lane = threadIdx.x & 31;
    if (node >= n) return;
    const float* hr = Hb + (size_t)node * HC;
    float h0a = hr[lane], h0b = hr[lane + 32];
    float h1a = hr[lane + 64], h1b = hr[lane + 96];
    float s0 = h0a * as_[lane]      + h0b * as_[lane + 32];
    float s1 = h1a * as_[lane + 64] + h1b * as_[lane + 96];
    float d0 = h0a * ad_[lane]      + h0b * ad_[lane + 32];
    float d1 = h1a * ad_[lane + 64] + h1b * ad_[lane + 96];
    #pragma unroll
    for (int off = 16; off > 0; off >>= 1) {
        s0 += __shfl_down(s0, off);
        s1 += __shfl_down(s1, off);
        d0 += __shfl_down(d0, off);
        d1 += __shfl_down(d1, off);
    }
    if (lane == 0) {
        oas[node * 2 + 0] = s0; oas[node * 2 + 1] = s1;
        oad[node * 2 + 0] = d0; oad[node * 2 + 1] = d1;
    }
}

// ---------------------------------------------------------------------------
// CSR construction (once per launch; edge topology is layer-invariant)
// ---------------------------------------------------------------------------
__device__ __forceinline__ void edge_ends(int e, const int* __restrict__ ei,
                                          int& s, int& d)
{
    if (e < NEDGES) { s = ei[e]; d = ei[NEDGES + e]; }
    else            { s = d = e - NEDGES; }          // self loops
}

__global__ void csr_zero(int* __restrict__ deg, int n)
{
    const int i = blockIdx.x * blockDim.x + threadIdx.x;
    if (i < n) deg[i] = 0;
}

__global__ void csr_count(const int* __restrict__ ei, int* __restrict__ deg)
{
    const int e = blockIdx.x * blockDim.x + threadIdx.x;
    if (e >= E2) return;
    int s, d; edge_ends(e, ei, s, d);
    atomicAdd(&deg[d], 1);
}

// block-level exclusive scan (Hillis-Steele in shared), 256 elems/block
__global__ void scan1(const int* __restrict__ deg, int* __restrict__ offs,
                      int* __restrict__ bsum, int n)
{
    __shared__ int sh[256];
    const int t = threadIdx.x;
    const int i = blockIdx.x * 256 + t;
    int x = (i < n) ? deg[i] : 0;
    sh[t] = x;
    __syncthreads();
    #pragma unroll
    for (int s = 1; s < 256; s <<= 1) {
        int v = (t >= s) ? sh[t - s] : 0;
        __syncthreads();
        sh[t] += v;
        __syncthreads();
    }
    if (i < n) offs[i] = sh[t] - x;          // exclusive within block
    if (t == 255) bsum[blockIdx.x] = sh[255];
}

__global__ void scan2(int* __restrict__ bsum, int nb)
{
    __shared__ int sh[256];
    const int t = threadIdx.x;
    int x = (t < nb) ? bsum[t] : 0;
    sh[t] = x;
    __syncthreads();
    #pragma unroll
    for (int s = 1; s < 256; s <<= 1) {
        int v = (t >= s) ? sh[t - s] : 0;
        __syncthreads();
        sh[t] += v;
        __syncthreads();
    }
    if (t < nb) bsum[t] = sh[t] - x;         // exclusive block offsets
}

__global__ void scan3(int* __restrict__ offs, int* __restrict__ cursor,
                      const int* __restrict__ bsum, int n)
{
    const int i = blockIdx.x * 256 + threadIdx.x;
    if (i >= n) return;
    const int o = offs[i] + bsum[blockIdx.x];
    offs[i] = o;
    cursor[i] = o;
}

__global__ void csr_fill(const int* __restrict__ ei, int* __restrict__ cursor,
                         int* __restrict__ csr_src)
{
    const int e = blockIdx.x * blockDim.x + threadIdx.x;
    if (e >= E2) return;
    int s, d; edge_ends(e, ei, s, d);
    const int pos = atomicAdd(&cursor[d], 1);
    csr_src[pos] = s;
}

// ---------------------------------------------------------------------------
// Fused per-dst softmax + aggregation + bias + ELU.  One wave32 per dst node.
// Phase A: lane-parallel max / sum-exp over incoming edges (butterfly shfl).
// Phase B: serial edge loop; each lane holds 4 channels (float4 acc in regs),
//          h[src] rows gathered as coalesced b128 loads. No global atomics.
// ---------------------------------------------------------------------------
__global__ __launch_bounds__(256) void gat_node_agg(
    const int* __restrict__ offs, const int* __restrict__ deg,
    const int* __restrict__ csr_src,
    const float* __restrict__ asrc, const float* __restrict__ adst,
    const float* __restrict__ hbuf, const float* __restrict__ bias,
    float* __restrict__ xbuf, int n)
{
    const int node = blockIdx.x * (blockDim.x >> 5) + (threadIdx.x >> 5);
    const int lane = threadIdx.x & 31;
    if (node >= n) return;

    const int off = offs[node];
    const int dg  = deg[node];                 // >= 1 (self loop)
    const float ad0 = adst[node * 2 + 0];
    const float ad1 = adst[node * 2 + 1];

    // ---- max over edges (both heads) ----
    float m0 = -__builtin_inff(), m1 = -__builtin_inff();
    for (int i = lane; i < dg; i += 32) {
        const int s = csr_src[off + i];
        float v0 = asrc[s * 2 + 0] + ad0;  v0 = v0 > 0.f ? v0 : NEG_SLOPE * v0;
        float v1 = asrc[s * 2 + 1] + ad1;  v1 = v1 > 0.f ? v1 : NEG_SLOPE * v1;
        m0 = fmaxf(m0, v0); m1 = fmaxf(m1, v1);
    }
    #pragma unroll
    for (int w = 16; w > 0; w >>= 1) {
        m0 = fmaxf(m0, __shfl_xor(m0, w));
        m1 = fmaxf(m1, __shfl_xor(m1, w));
    }

    // ---- sum of exp ----
    float z0 = 0.f, z1 = 0.f;
    for (int i = lane; i < dg; i += 32) {
        const int s = csr_src[off + i];
        float v0 = asrc[s * 2 + 0] + ad0;  v0 = v0 > 0.f ? v0 : NEG_SLOPE * v0;
        float v1 = asrc[s * 2 + 1] + ad1;  v1 = v1 > 0.f ? v1 : NEG_SLOPE * v1;
        z0 += __expf(v0 - m0); z1 += __expf(v1 - m1);
    }
    #pragma unroll
    for (int w = 16; w > 0; w >>= 1) {
        z0 += __shfl_xor(z0, w);
        z1 += __shfl_xor(z1, w);
    }

    // ---- accumulate messages: lane owns channels [4*lane, 4*lane+4) ----
    const int h   = lane >> 4;                         // head of my channels
    const float mh  = h ? m1 : m0;
    const float rzh = 1.f / (h ? z1 : z0);
    const float adh = h ? ad1 : ad0;
    float4 acc = make_float4(0.f, 0.f, 0.f, 0.f);
    for (int j = 0; j < dg; ++j) {
        const int s = csr_src[off + j];                // uniform across wave
        float v = asrc[s * 2 + h] + adh;
        v = v > 0.f ? v : NEG_SLOPE * v;
        const float al = __expf(v - mh) * rzh;
        const float4 hh = *reinterpret_cast<const float4*>(
            hbuf + (size_t)s * HC + lane * 4);
        acc.x += hh.x * al; acc.y += hh.y * al;
        acc.z += hh.z * al; acc.w += hh.w * al;
    }

    // ---- bias + ELU + store ----
    const float4 bb = *reinterpret_cast<const float4*>(bias + lane * 4);
    float4 o;
    o.x = acc.x + bb.x; o.x = o.x > 0.f ? o.x : __expf(o.x) - 1.f;
    o.y = acc.y + bb.y; o.y = o.y > 0.f ? o.y : __expf(o.y) - 1.f;
    o.z = acc.z + bb.z; o.z = o.z > 0.f ? o.z : __expf(o.z) - 1.f;
    o.w = acc.w + bb.w; o.w = o.w > 0.f ? o.w : __expf(o.w) - 1.f;
    *reinterpret_cast<float4*>(xbuf + (size_t)node * HC + lane * 4) = o;
}

// ---------------------------------------------------------------------------
// Global mean pool + linear + softmax head
// ---------------------------------------------------------------------------
__global__ void pool_init(float* __restrict__ pooled, float* __restrict__ cnt)
{
    const int tid = blockIdx.x * blockDim.x + threadIdx.x;
    if (tid < NGRAPH * HC) pooled[tid] = 0.0f;
    if (tid < NGRAPH) cnt[tid] = 0.0f;
}

__global__ void pool_sum(const float* __restrict__ xbuf, const int* __restrict__ batch,
                         float* __restrict__ pooled, float* __restrict__ cnt, int n)
{
    const int tid = blockIdx.x * blockDim.x + threadIdx.x;
    if (tid >= n * HC) return;
    const int node = tid >> 7;
    const int c = tid & (HC - 1);
    const int g = batch[node];
    atomicAdd(&pooled[(size_t)g * HC + c], xbuf[tid]);
    if (c == 0) atomicAdd(&cnt[g], 1.0f);
}

__global__ void gat_head(const float* __restrict__ pooled, const float* __restrict__ cnt,
                         const float* __restrict__ lw, const float* __restrict__ lb,
                         float* __restrict__ out)
{
    __shared__ float lg[16];
    const int g = blockIdx.x;
    const int t = threadIdx.x;
    const float inv = 1.0f / fmaxf(cnt[g], 1.0f);
    if (t < NCLS) {
        float acc = lb[t];
        for (int k = 0; k < HC; ++k)
            acc += pooled[(size_t)g * HC + k] * inv * lw[k * NCLS + t];
        lg[t] = acc;
    }
    __syncthreads();
    if (t == 0) {
        float mx = lg[0];
        for (int k = 1; k < NCLS; ++k) mx = fmaxf(mx, lg[k]);
        float ex[NCLS]; float sum = 0.0f;
        for (int k = 0; k < NCLS; ++k) { ex[k] = __expf(lg[k] - mx); sum += ex[k]; }
        for (int k = 0; k < NCLS; ++k) out[g * NCLS + k] = ex[k] / sum;
    }
}

extern "C" void kernel_launch(void* const* d_in, const int* in_sizes, int n_in,
                              void* d_out, int out_size, void* d_ws, size_t ws_size,
                              hipStream_t stream)
{
    const float* x     = (const float*)d_in[0];
    const int*   ei    = (const int*)d_in[1];
    const int*   batch = (const int*)d_in[2];
    const float* W     = (const float*)d_in[3];   // [3,128,128]
    const float* att_s = (const float*)d_in[4];   // [3,2,64]
    const float* att_d = (const float*)d_in[5];
    const float* bias  = (const float*)d_in[6];   // [3,128]
    const float* lw    = (const float*)d_in[7];   // [128,10]
    const float* lb    = (const float*)d_in[8];   // [10]
    float* out = (float*)d_out;

    // ---- carve workspace: floats first, then ints ----
    float* wf = (float*)d_ws;
    size_t o = 0;
    float* xbuf   = wf + o; o += (size_t)NNODES * HC;
    float* hbuf   = wf + o; o += (size_t)NNODES * HC;
    float* asrc   = wf + o; o += (size_t)NNODES * NHEAD;
    float* adst   = wf + o; o += (size_t)NNODES * NHEAD;
    float* pooled = wf + o; o += (size_t)NGRAPH * HC;
    float* cnt    = wf + o; o += (size_t)NGRAPH;
    int* wi = (int*)(wf + o);
    size_t oi = 0;
    int* deg     = wi + oi; oi += NNODES;
    int* offs    = wi + oi; oi += NNODES;
    int* cursor  = wi + oi; oi += NNODES;
    int* bsum    = wi + oi; oi += 256;
    int* csr_src = wi + oi; oi += E2;

    const dim3 blk(256);
    const unsigned gN     = (NNODES + 255) / 256;        // 196
    const unsigned gEdge  = (E2 + 255) / 256;
    const unsigned gNode  = (NNODES + 7) / 8;            // 1 wave / node
    const unsigned gElem  = ((unsigned)NNODES * HC + 255) / 256;

    // ---- build dst-CSR once (layer-invariant topology) ----
    csr_zero<<<gN, blk, 0, stream>>>(deg, NNODES);
    csr_count<<<gEdge, blk, 0, stream>>>(ei, deg);
    scan1<<<gN, blk, 0, stream>>>(deg, offs, bsum, NNODES);
    scan2<<<1, blk, 0, stream>>>(bsum, (int)gN);
    scan3<<<gN, blk, 0, stream>>>(offs, cursor, bsum, NNODES);
    csr_fill<<<gEdge, blk, 0, stream>>>(ei, cursor, csr_src);

    // ---- 3 GAT layers ----
    for (int l = 0; l < 3; ++l) {
        const float* xin = (l == 0) ? x : xbuf;
        gat_gemm_wmma<<<NNODES / 16, blk, 0, stream>>>(
            xin, W + (size_t)l * HC * HC, hbuf, NNODES);
        gat_alpha<<<gNode, blk, 0, stream>>>(
            hbuf, att_s + l * HC, att_d + l * HC, asrc, adst, NNODES);
        gat_node_agg<<<gNode, blk, 0, stream>>>(
            offs, deg, csr_src, asrc, adst, hbuf, bias + l * HC, xbuf, NNODES);
    }

    // ---- mean pool + head ----
    pool_init<<<(NGRAPH * HC + 255) / 256, blk, 0, stream>>>(pooled, cnt);
    pool_sum<<<gElem, blk, 0, stream>>>(xbuf, batch, pooled, cnt, NNODES);
    gat_head<<<NGRAPH, dim3(32), 0, stream>>>(pooled, cnt, lw, lb, out);
}